// SpatialTemporalGNN_85839216378483
// MI455X (gfx1250) — compile-verified
//
#include <hip/hip_runtime.h>
#include <hip/hip_bf16.h>
#include <math.h>

// ---------------------------------------------------------------------------
// SpatialTemporalGNN for MI455X (gfx1250, wave32, WMMA)
//   GCN(32->64->32->16) + mean-pool + 2-layer biLSTM(H=128,T=24,B=64) + MLP
// All dense GEMMs use v_wmma_f32_16x16x4_f32 (fp32 path: problem is
// bandwidth/latency bound, so fp32 WMMA gives reference precision for free).
// ---------------------------------------------------------------------------

typedef __attribute__((ext_vector_type(2))) float v2f;
typedef __attribute__((ext_vector_type(8))) float v8f;

static __device__ __forceinline__ v8f wmma_k4(v8f acc, v2f a, v2f b) {
  // D = A(16x4,f32) * B(4x16,f32) + C ; 8 args: neg_a,A,neg_b,B,c_mod,C,reuse_a,reuse_b
  return __builtin_amdgcn_wmma_f32_16x16x4_f32(false, a, false, b, (short)0, acc,
                                               false, false);
}

// ---------------------------------------------------------------------------
// Utility kernels
// ---------------------------------------------------------------------------
__global__ void fill_kernel(float* __restrict__ p, float v, size_t n) {
  size_t i = (size_t)blockIdx.x * blockDim.x + threadIdx.x;
  if (i < n) p[i] = v;
}

__global__ void deg_edges_kernel(const int* __restrict__ dst,
                                 float* __restrict__ deg, int E) {
  int e = blockIdx.x * blockDim.x + threadIdx.x;
  if (e < E) atomicAdd(&deg[dst[e]], 1.0f);
}

__global__ void rsqrt_kernel(const float* __restrict__ deg,
                             float* __restrict__ dis, int n) {
  int i = blockIdx.x * blockDim.x + threadIdx.x;
  if (i < n) dis[i] = rsqrtf(deg[i]);
}

// ---------------------------------------------------------------------------
// Generic fp32 WMMA GEMM:  C[M,N] = act( A[M,K] @ W[N,K]^T + bias + bias2 )
// One 16x16 output tile per wave, K-loop in steps of 4 via wmma 16x16x4.
// Requires K % 4 == 0, N % 16 == 0 (true for every call site here).
// ---------------------------------------------------------------------------
__global__ __launch_bounds__(256)
void gemm_wmma_kernel(const float* __restrict__ A, const float* __restrict__ W,
                      const float* __restrict__ bias,
                      const float* __restrict__ bias2,
                      float* __restrict__ C, int M, int N, int K, int relu) {
  const int lane = threadIdx.x & 31;
  const int wave = threadIdx.x >> 5;
  const int tile = blockIdx.x * (blockDim.x >> 5) + wave;
  const int ntilesN = N >> 4;
  const int mt = tile / ntilesN;
  const int nt = tile - mt * ntilesN;
  if (mt * 16 >= M) return;  // wave-uniform: EXEC stays all-ones for WMMA

  const int half = lane >> 4;   // 0: K pair {0,1}; 1: K pair {2,3}
  const int l    = lane & 15;
  int row = mt * 16 + l; if (row > M - 1) row = M - 1;
  const int col = nt * 16 + l;
  const float* __restrict__ Arow = A + (size_t)row * K;
  const float* __restrict__ Wrow = W + (size_t)col * K;

  v8f acc = {};
  for (int k0 = 0; k0 < K; k0 += 4) {
    const int ka = k0 + half * 2;
    v2f a, b;
    a.x = Arow[ka]; a.y = Arow[ka + 1];
    b.x = Wrow[ka]; b.y = Wrow[ka + 1];
    acc = wmma_k4(acc, a, b);
  }

  float bb = 0.0f;
  if (bias)  bb += bias[col];
  if (bias2) bb += bias2[col];
#pragma unroll
  for (int j = 0; j < 8; ++j) {
    const int r = mt * 16 + j + half * 8;  // VGPR j -> rows j / j+8
    if (r < M) {
      float v = acc[j] + bb;
      if (relu) v = fmaxf(v, 0.0f);
      C[(size_t)r * N + col] = v;
    }
  }
}

// ---------------------------------------------------------------------------
// GCN edge scatter: agg[dst] += hw[src] * dis[src]*dis[dst]   (float4 chunks)
// ---------------------------------------------------------------------------
__global__ void edge_scatter_kernel(const int* __restrict__ src,
                                    const int* __restrict__ dst,
                                    const float* __restrict__ dis,
                                    const float* __restrict__ hw,
                                    float* __restrict__ agg,
                                    long long total, int F, int lc) {
  long long tid = (long long)blockIdx.x * blockDim.x + threadIdx.x;
  if (tid >= total) return;
  const long long e = tid >> lc;
  const int c = (int)(tid & ((1 << lc) - 1));
  const int s = src[e], d = dst[e];
  const float coef = dis[s] * dis[d];
  const float4 v = ((const float4*)(hw + (size_t)s * F))[c];
  float* ap = agg + (size_t)d * F + (size_t)c * 4;
  atomicAdd(ap + 0, v.x * coef);
  atomicAdd(ap + 1, v.y * coef);
  atomicAdd(ap + 2, v.z * coef);
  atomicAdd(ap + 3, v.w * coef);
}

// h_out = relu(agg + hw*dis^2 + bias)
__global__ void gcn_finish_kernel(const float* __restrict__ agg,
                                  const float* __restrict__ hw,
                                  const float* __restrict__ dis,
                                  const float* __restrict__ bias,
                                  float* __restrict__ out, int N, int F) {
  size_t i = (size_t)blockIdx.x * blockDim.x + threadIdx.x;
  if (i >= (size_t)N * F) return;
  const int n = (int)(i / F);
  const int f = (int)(i - (size_t)n * F);
  const float d = dis[n];
  out[i] = fmaxf(agg[i] + hw[i] * d * d + bias[f], 0.0f);
}

// ---------------------------------------------------------------------------
// Global mean pool (F fixed = 16)
// ---------------------------------------------------------------------------
__global__ void pool_scatter_kernel(const float* __restrict__ h,
                                    const int* __restrict__ batch,
                                    float* __restrict__ sums,
                                    float* __restrict__ cnt, int N) {
  int n = blockIdx.x * blockDim.x + threadIdx.x;
  if (n >= N) return;
  const int b = batch[n];
  atomicAdd(&cnt[b], 1.0f);
#pragma unroll
  for (int f = 0; f < 16; ++f) atomicAdd(&sums[b * 16 + f], h[(size_t)n * 16 + f]);
}

__global__ void pool_finish_kernel(float* __restrict__ g,
                                   const float* __restrict__ cnt, int Bsz) {
  int i = blockIdx.x * blockDim.x + threadIdx.x;
  if (i >= Bsz * 16) return;
  g[i] = g[i] / fmaxf(cnt[i >> 4], 1.0f);
}

// ---------------------------------------------------------------------------
// Persistent single-workgroup LSTM direction kernel.
//  h(32KB), c(32KB), gates(128KB) live in LDS  -> 192KB (CDNA5 WGP has 320KB).
//  Per step: gates = pre[t] + h @ Whh^T via wmma f32 16x16x4 (A read from LDS),
//  then the elementwise cell update; 8 waves / 256 threads.
// ---------------------------------------------------------------------------
#define LT 24
#define LB 64
#define LH 128
#define LG 512  // 4*H

__global__ __launch_bounds__(256)
void lstm_dir_kernel(const float* __restrict__ pre, int preTStride,
                     const float* __restrict__ Whh,
                     float* __restrict__ out, int outStride, int outOff,
                     int reverse) {
  extern __shared__ float smem[];
  float* h_s = smem;                  // LB*LH
  float* c_s = h_s + LB * LH;         // LB*LH
  float* g_s = c_s + LB * LH;         // LB*LG

  const int tid  = threadIdx.x;
  const int lane = tid & 31;
  const int wave = tid >> 5;
  const int half = lane >> 4;
  const int l    = lane & 15;

  for (int i = tid; i < LB * LH; i += 256) { h_s[i] = 0.0f; c_s[i] = 0.0f; }
  __syncthreads();

  for (int step = 0; step < LT; ++step) {
    const int t = reverse ? (LT - 1 - step) : step;
    const float* __restrict__ preT = pre + (size_t)t * preTStride;

    // gates = preT + h @ Whh^T : (64x512), tiles 4x32, 16 tiles per wave
    for (int tt = wave; tt < (LB / 16) * (LG / 16); tt += 8) {
      const int mt = tt >> 5;
      const int nt = tt & 31;
      const float* __restrict__ hrow = h_s + (mt * 16 + l) * LH;
      const float* __restrict__ wrow = Whh + (size_t)(nt * 16 + l) * LH;
      v8f acc = {};
      for (int k0 = 0; k0 < LH; k0 += 4) {
        const int ka = k0 + half * 2;
        v2f a, b;
        a.x = hrow[ka]; a.y = hrow[ka + 1];
        b.x = wrow[ka]; b.y = wrow[ka + 1];
        acc = wmma_k4(acc, a, b);
      }
#pragma unroll
      for (int j = 0; j < 8; ++j) {
        const int r = mt * 16 + j + half * 8;
        const int cc = nt * 16 + l;
        g_s[r * LG + cc] = acc[j] + preT[r * LG + cc];
      }
    }
    __syncthreads();

    // elementwise LSTM cell update + emit h_t
    for (int e = tid; e < LB * LH; e += 256) {
      const int b = e >> 7;
      const int j = e & 127;
      const float gi = g_s[b * LG + j];
      const float gf = g_s[b * LG + 128 + j];
      const float gg = g_s[b * LG + 256 + j];
      const float go = g_s[b * LG + 384 + j];
      const float si = 1.0f / (1.0f + expf(-gi));
      const float sf = 1.0f / (1.0f + expf(-gf));
      const float so = 1.0f / (1.0f + expf(-go));
      const float c2 = sf * c_s[e] + si * tanhf(gg);
      const float h2 = so * tanhf(c2);
      c_s[e] = c2;
      h_s[e] = h2;
      out[((size_t)t * LB + b) * outStride + outOff + j] = h2;
    }
    __syncthreads();
  }
}

// final fusion layer: out[b] = m3[b,:] . W[0,:] + bias
__global__ void final_dot_kernel(const float* __restrict__ m3,
                                 const float* __restrict__ W,
                                 const float* __restrict__ b,
                                 float* __restrict__ out, int Bsz, int K) {
  int i = blockIdx.x * blockDim.x + threadIdx.x;
  if (i >= Bsz) return;
  float acc = b[0];
  for (int k = 0; k < K; ++k) acc += m3[(size_t)i * K + k] * W[k];
  out[i] = acc;
}

// ---------------------------------------------------------------------------
// Host-side orchestration
// ---------------------------------------------------------------------------
static inline void launch_gemm(const float* A, const float* W, const float* bias,
                               const float* bias2, float* C, int M, int N, int K,
                               int relu, hipStream_t stream) {
  const int tiles = ((M + 15) / 16) * (N / 16);
  const int blocks = (tiles + 7) / 8;  // 8 waves (16x16 tiles) per 256-thread block
  gemm_wmma_kernel<<<blocks, 256, 0, stream>>>(A, W, bias, bias2, C, M, N, K, relu);
}

extern "C" void kernel_launch(void* const* d_in, const int* in_sizes, int n_in,
                              void* d_out, int out_size, void* d_ws, size_t ws_size,
                              hipStream_t stream) {
  (void)in_sizes; (void)n_in; (void)out_size; (void)ws_size;
  constexpr int N = 100000, E = 1600000, Bsz = 64;

  // ---- inputs (setup_inputs dict order, tuples flattened in order) ----
  const float* x     = (const float*)d_in[0];
  const int*   ei    = (const int*)d_in[1];   // [2,E]: src=ei[0:E], dst=ei[E:2E]
  const int*   batch = (const int*)d_in[2];
  // d_in[3] = temporal_sequence (unused by the reference computation)
  const float* gW0 = (const float*)d_in[4];  const float* gb0 = (const float*)d_in[5];
  const float* gW1 = (const float*)d_in[6];  const float* gb1 = (const float*)d_in[7];
  const float* gW2 = (const float*)d_in[8];  const float* gb2 = (const float*)d_in[9];
  // lstm_w: [(fwd,rev) x 2 layers], each = (Wih, Whh, bih, bhh)
  const float* Wih0f = (const float*)d_in[10]; const float* Whh0f = (const float*)d_in[11];
  const float* bih0f = (const float*)d_in[12]; const float* bhh0f = (const float*)d_in[13];
  const float* Wih0r = (const float*)d_in[14]; const float* Whh0r = (const float*)d_in[15];
  const float* bih0r = (const float*)d_in[16]; const float* bhh0r = (const float*)d_in[17];
  const float* Wih1f = (const float*)d_in[18]; const float* Whh1f = (const float*)d_in[19];
  const float* bih1f = (const float*)d_in[20]; const float* bhh1f = (const float*)d_in[21];
  const float* Wih1r = (const float*)d_in[22]; const float* Whh1r = (const float*)d_in[23];
  const float* bih1r = (const float*)d_in[24]; const float* bhh1r = (const float*)d_in[25];
  const float* fW0 = (const float*)d_in[26]; const float* fb0 = (const float*)d_in[27];
  const float* fW1 = (const float*)d_in[28]; const float* fb1 = (const float*)d_in[29];
  const float* fW2 = (const float*)d_in[30]; const float* fb2 = (const float*)d_in[31];
  const float* fW3 = (const float*)d_in[32]; const float* fb3 = (const float*)d_in[33];

  const int* src = ei;
  const int* dst = ei + E;

  // ---- workspace layout (floats) ----
  float* ws = (float*)d_ws;
  size_t off = 0;
  float* deg  = ws + off; off += N;
  float* dis  = ws + off; off += N;
  float* hw   = ws + off; off += (size_t)N * 64;   // per-layer H @ W^T (reused)
  float* agg  = ws + off; off += (size_t)N * 64;   // per-layer edge aggregate (reused)
  float* hcur = ws + off; off += (size_t)N * 64;   // per-layer activations (reused)
  float* g    = ws + off; off += Bsz * 16;         // pooled graph embedding
  float* cnt  = ws + off; off += Bsz;
  float* pre1f = ws + off; off += Bsz * LG;        // time-invariant layer-1 input proj
  float* pre1r = ws + off; off += Bsz * LG;
  float* pre2f = ws + off; off += (size_t)LT * Bsz * LG;
  float* pre2r = ws + off; off += (size_t)LT * Bsz * LG;
  float* xs2  = ws + off; off += (size_t)LT * Bsz * 256;  // layer-1 biLSTM outputs
  float* xs3  = ws + off; off += (size_t)LT * Bsz * 256;  // layer-2 biLSTM outputs
  float* m1   = ws + off; off += Bsz * 256;
  float* m2   = ws + off; off += Bsz * 128;
  float* m3   = ws + off; off += Bsz * 64;

  // ---- degrees & symmetric normalization ----
  fill_kernel<<<(N + 255) / 256, 256, 0, stream>>>(deg, 1.0f, (size_t)N);
  deg_edges_kernel<<<(E + 255) / 256, 256, 0, stream>>>(dst, deg, E);
  rsqrt_kernel<<<(N + 255) / 256, 256, 0, stream>>>(deg, dis, N);

  // ---- GCN layers: hw = h@W^T ; agg = scatter(msg) ; h = relu(agg+hw*dis^2+b)
  const float* hin = x;
  const float* Ws[3]   = {gW0, gW1, gW2};
  const float* bs[3]   = {gb0, gb1, gb2};
  const int    Fin[3]  = {32, 64, 32};
  const int    Fout[3] = {64, 32, 16};
  for (int L = 0; L < 3; ++L) {
    const int F = Fout[L];
    launch_gemm(hin, Ws[L], nullptr, nullptr, hw, N, F, Fin[L], 0, stream);
    fill_kernel<<<((size_t)N * F + 255) / 256, 256, 0, stream>>>(agg, 0.0f, (size_t)N * F);
    const int lc = (F == 64) ? 4 : (F == 32) ? 3 : 2;  // log2(F/4)
    const long long total = (long long)E << lc;
    edge_scatter_kernel<<<(int)((total + 255) / 256), 256, 0, stream>>>(
        src, dst, dis, hw, agg, total, F, lc);
    gcn_finish_kernel<<<((size_t)N * F + 255) / 256, 256, 0, stream>>>(
        agg, hw, dis, bs[L], hcur, N, F);
    hin = hcur;
  }

  // ---- global mean pool -> g [64,16] ----
  fill_kernel<<<(Bsz * 16 + 255) / 256, 256, 0, stream>>>(g, 0.0f, (size_t)Bsz * 16);
  fill_kernel<<<1, 256, 0, stream>>>(cnt, 0.0f, (size_t)Bsz);
  pool_scatter_kernel<<<(N + 255) / 256, 256, 0, stream>>>(hcur, batch, g, cnt, N);
  pool_finish_kernel<<<(Bsz * 16 + 255) / 256, 256, 0, stream>>>(g, cnt, Bsz);

  // ---- biLSTM layer 1 (input is g broadcast over T: projection is t-invariant)
  launch_gemm(g, Wih0f, bih0f, bhh0f, pre1f, Bsz, LG, 16, 0, stream);
  launch_gemm(g, Wih0r, bih0r, bhh0r, pre1r, Bsz, LG, 16, 0, stream);
  const size_t lstm_smem = (size_t)(2 * LB * LH + LB * LG) * sizeof(float);  // 192KB
  lstm_dir_kernel<<<1, 256, lstm_smem, stream>>>(pre1f, 0, Whh0f, xs2, 256, 0,   0);
  lstm_dir_kernel<<<1, 256, lstm_smem, stream>>>(pre1r, 0, Whh0r, xs2, 256, 128, 1);

  // ---- biLSTM layer 2 (batched input projection over all T) ----
  launch_gemm(xs2, Wih1f, bih1f, bhh1f, pre2f, LT * Bsz, LG, 256, 0, stream);
  launch_gemm(xs2, Wih1r, bih1r, bhh1r, pre2r, LT * Bsz, LG, 256, 0, stream);
  lstm_dir_kernel<<<1, 256, lstm_smem, stream>>>(pre2f, LB * LG, Whh1f, xs3, 256, 0,   0);
  lstm_dir_kernel<<<1, 256, lstm_smem, stream>>>(pre2r, LB * LG, Whh1r, xs3, 256, 128, 1);

  // ---- fusion MLP on last timestep [64,256] ----
  const float* last = xs3 + (size_t)(LT - 1) * Bsz * 256;
  launch_gemm(last, fW0, fb0, nullptr, m1, Bsz, 256, 256, 1, stream);
  launch_gemm(m1,  fW1, fb1, nullptr, m2, Bsz, 128, 256, 1, stream);
  launch_gemm(m2,  fW2, fb2, nullptr, m3, Bsz, 64, 128, 1, stream);
  final_dot_kernel<<<1, 64, 0, stream>>>(m3, fW3, fb3, (float*)d_out, Bsz, 64);
}